// LearnableMask_47983374630951
// MI455X (gfx1250) — compile-verified
//
#include <hip/hip_runtime.h>

// ---------------------------------------------------------------------------
// LearnableMask on MI455X (gfx1250)
//
// Roofline: 268 MB of traffic, 0.034 GFLOP -> 0.125 FLOP/B, memory-bound,
// ~11.5 us floor at 23.3 TB/s. No WMMA (diag-matmul would be 1024x the FLOPs
// for an identical result). Data path: async global->LDS staging of the 4KB
// mask (ASYNCcnt path), then b128 non-temporal streaming multiply.
//
// apply_mask_kernel is defined FIRST so the disasm snippet shows the
// global_load_async_to_lds_b128 / s_wait_asynccnt / NT b128 stream.
// ---------------------------------------------------------------------------

typedef float v4f __attribute__((ext_vector_type(4)));
typedef int   v4i __attribute__((ext_vector_type(4)));

#define MASK_DIM   1024
#define F4_PER_D   (MASK_DIM / 4)   // 256 float4 groups per channel row
#define TPB        256              // 8 wave32 per block
#define UNROLL     8                // 8 x float4 = 128B per thread each way

// ---------------------------------------------------------------------------
// Kernel B: out[j] = x[j] * mask[j % DIM], streamed as float4 / b128.
// Block mapping is contiguous: f4 index j = blk*2048 + k*256 + tid, so
// (j % 256) == tid and each thread needs exactly one mask float4, which the
// block stages in LDS via the gfx1250 async global->LDS path.
// ---------------------------------------------------------------------------
__global__ __launch_bounds__(TPB) void apply_mask_kernel(
    const float* __restrict__ x,
    const float* __restrict__ mask,   // MASK_DIM floats (already computed)
    float* __restrict__ out,
    long long nf4)
{
    __shared__ v4f smask[F4_PER_D];   // 4 KB
    const unsigned tid = threadIdx.x;

    // ---- stage mask into LDS: one b128 per lane, async (ASYNCcnt) ----
    {
        const v4f* gsrc = ((const v4f*)mask) + tid;
#if defined(__AMDGCN__)
#if __has_builtin(__builtin_amdgcn_global_load_async_to_lds_b128)
        __builtin_amdgcn_global_load_async_to_lds_b128(
            (__attribute__((address_space(1))) v4i*)gsrc,
            (__attribute__((address_space(3))) v4i*)&smask[tid],
            /*offset=*/0, /*cpol=*/0);
#else
        unsigned lds_off = (unsigned)(unsigned long long)
            ((__attribute__((address_space(3))) v4f*)&smask[tid]);
        asm volatile("global_load_async_to_lds_b128 %0, %1, off"
                     :: "v"(lds_off), "v"(gsrc)
                     : "memory");
#endif
#if __has_builtin(__builtin_amdgcn_s_wait_asynccnt)
        __builtin_amdgcn_s_wait_asynccnt(0);
#else
        asm volatile("s_wait_asynccnt 0" ::: "memory");
#endif
#else
        // host-pass placeholder (never executed on device)
        smask[tid] = gsrc[0];
#endif
    }
    __syncthreads();

    const v4f mf = smask[tid];        // tid < 256 == F4_PER_D

    const long long per_block = (long long)(TPB * UNROLL);
    const long long base = (long long)blockIdx.x * per_block + (long long)tid;

    if (base + (long long)(UNROLL - 1) * TPB < nf4) {
        // Full block (always true for the reference shape): no per-iter guard.
#pragma unroll
        for (int k = 0; k < UNROLL; ++k) {
            const long long j = base + (long long)k * TPB;
            v4f xv = __builtin_nontemporal_load(((const v4f*)x) + j);
            __builtin_nontemporal_store(xv * mf, ((v4f*)out) + j);
        }
    } else {
        // Tail block for general sizes.
#pragma unroll
        for (int k = 0; k < UNROLL; ++k) {
            const long long j = base + (long long)k * TPB;
            if (j < nf4) {
                v4f xv = __builtin_nontemporal_load(((const v4f*)x) + j);
                __builtin_nontemporal_store(xv * mf, ((v4f*)out) + j);
            }
        }
    }
}

// ---------------------------------------------------------------------------
// Kernel A: mask = DIM * softmax(mask_param) * _mask   (single 1024-thread WG)
// LDS tree down to 32 partials, then wave32 shuffle reduction.
// ---------------------------------------------------------------------------
__global__ __launch_bounds__(MASK_DIM) void mask_softmax_kernel(
    const float* __restrict__ mask_param,
    const float* __restrict__ mask_fixed,
    float* __restrict__ mask_out)
{
    __shared__ float red[MASK_DIM];
    __shared__ float bcast;
    const int tid = threadIdx.x;

    const float v = mask_param[tid];

    // ---- max-reduce ----
    red[tid] = v;
    __syncthreads();
    for (int s = MASK_DIM / 2; s >= 32; s >>= 1) {
        if (tid < s) red[tid] = fmaxf(red[tid], red[tid + s]);
        __syncthreads();
    }
    if (tid < 32) {                    // one wave32: shuffle reduce
        float m = red[tid];
        for (int off = 16; off > 0; off >>= 1)
            m = fmaxf(m, __shfl_xor(m, off, 32));
        if (tid == 0) bcast = m;
    }
    __syncthreads();
    const float mx = bcast;
    __syncthreads();

    const float e = __expf(v - mx);

    // ---- sum-reduce ----
    red[tid] = e;
    __syncthreads();
    for (int s = MASK_DIM / 2; s >= 32; s >>= 1) {
        if (tid < s) red[tid] += red[tid + s];
        __syncthreads();
    }
    if (tid < 32) {
        float sum = red[tid];
        for (int off = 16; off > 0; off >>= 1)
            sum += __shfl_xor(sum, off, 32);
        if (tid == 0) bcast = sum;
    }
    __syncthreads();
    const float inv_sum = 1.0f / bcast;

    mask_out[tid] = (float)MASK_DIM * e * inv_sum * mask_fixed[tid];
}

// ---------------------------------------------------------------------------
// Launch: outputs are (xm, mask) concatenated flat in d_out.
// ---------------------------------------------------------------------------
extern "C" void kernel_launch(void* const* d_in, const int* in_sizes, int n_in,
                              void* d_out, int out_size, void* d_ws, size_t ws_size,
                              hipStream_t stream) {
    const float* x          = (const float*)d_in[0];   // [8,4096,1024] fp32
    const float* mask_param = (const float*)d_in[1];   // [1024] fp32
    const float* mask_fixed = (const float*)d_in[2];   // [1024] fp32

    const long long n = (long long)in_sizes[0];        // 33,554,432
    float* out_x    = (float*)d_out;                   // xm, n floats
    float* out_mask = (float*)d_out + n;               // mask, 1024 floats

    // Kernel A: compute the mask once (also serves as kernel B's input).
    mask_softmax_kernel<<<1, MASK_DIM, 0, stream>>>(mask_param, mask_fixed, out_mask);

    // Kernel B: stream x * mask.
    const long long nf4       = n / 4;
    const long long per_block = (long long)TPB * UNROLL;
    const int grid = (int)((nf4 + per_block - 1) / per_block);  // 4096 blocks
    apply_mask_kernel<<<grid, TPB, 0, stream>>>(x, out_mask, out_x, nf4);
}